// SpikeMLP_31911607009779
// MI455X (gfx1250) — compile-verified
//
#include <hip/hip_runtime.h>

// ---------------------------------------------------------------------------
// Fused spiking-MLP kernel for MI455X (gfx1250, wave32, WMMA).
//
// Layer 1 (per workgroup = 16-batch tile, per wave = 16 hidden neurons):
//   A[b,o] = sum_i mask(x[b,i]) * W1[o,i]          (mask: x != 1.0)
//   B[b,o] = sum_i mask(x[b,i]) * x[b,i] * W1[o,i]
//   via V_WMMA_F32_16X16X4_F32, K = 800 (200 k-steps, 2 WMMA each).
//   mem(t) = A*t*DT - B  (exact for t >= 256 where all relus are active);
//   first t in [256,511] with mem > (511-t)*DT  ->  spike time h.
//
// Layer 2 (wave 0 of each workgroup, from LDS h-tile):
//   C[b,o] = sum_i mask(h) * W2[o,i],  D[b,o] = sum_i mask(h)*h * W2[o,i]
//   via WMMA, K = 128, N padded 10 -> 16; first tau = 1 + k*DT (k<1024)
//   with C*tau - D > 20  ->  output, else +INF.
// ---------------------------------------------------------------------------

typedef __attribute__((ext_vector_type(2))) float v2f;
typedef __attribute__((ext_vector_type(8))) float v8f;

#define SPIKE_DT (1.0f / 256.0f)

__launch_bounds__(256)
__global__ void spike_mlp_fused(const float* __restrict__ x,    // [128,800]
                                const float* __restrict__ W1,   // [128,800]
                                const float* __restrict__ W2,   // [10,128]
                                float* __restrict__ out) {      // [128,10]
  // h spike times for this workgroup's 16 batch rows, all 128 hidden neurons.
  // Stride 132 floats (132 % 64 == 4) keeps the layer-2 column reads
  // bank-conflict free across 16 lanes.
  __shared__ float hbuf[16][132];

  const int tid  = threadIdx.x;
  const int wave = tid >> 5;
  const int lane = tid & 31;
  const int r    = lane & 15;   // M index (A rows) / N index (B,C,D cols)
  const int half = lane >> 4;   // K sub-slot selector

  const int bBase = blockIdx.x * 16;  // batch tile
  const int oBase = wave * 16;        // hidden-neuron tile for this wave

  // ------------------------- Layer 1 GEMMs -------------------------
  v8f accA = {};  // mask    @ W1^T
  v8f accB = {};  // mask*x  @ W1^T

  const float* xrow = x  + (size_t)(bBase + r) * 800;
  const float* wrow = W1 + (size_t)(oBase + r) * 800;

  for (int k = 0; k < 800; k += 4) {
    const int kk = k + 2 * half;
    // A-matrix (16x4 f32): lanes 0-15 hold K = {0,1}, lanes 16-31 K = {2,3},
    // row M = lane%16.  B-matrix (4x16) mirrors with col N = lane%16.
    float x0 = xrow[kk];
    float x1 = xrow[kk + 1];
    v2f bf;  bf.x = wrow[kk];             bf.y = wrow[kk + 1];
    v2f am;  am.x = (x0 == 1.0f) ? 0.0f : 1.0f;
             am.y = (x1 == 1.0f) ? 0.0f : 1.0f;
    v2f ax;  ax.x = (x0 == 1.0f) ? 0.0f : x0;
             ax.y = (x1 == 1.0f) ? 0.0f : x1;
    accA = __builtin_amdgcn_wmma_f32_16x16x4_f32(false, am, false, bf,
                                                 (short)0, accA, false, false);
    accB = __builtin_amdgcn_wmma_f32_16x16x4_f32(false, ax, false, bf,
                                                 (short)0, accB, false, false);
  }

  // --------------- Layer 1 spike-time solve (per element) ---------------
  // C/D layout: acc[v] = D[m = v + 8*half][n = lane%16].
#pragma unroll
  for (int v = 0; v < 8; ++v) {
    float Am = accA[v];
    float Bm = accB[v];
    int idx = 511;  // vb[..., -1] is forced to 1 in the reference
#pragma unroll 1
    for (int t = 256; t < 511; ++t) {
      if (fmaf(Am, (float)t * SPIKE_DT, -Bm) > (float)(511 - t) * SPIKE_DT) {
        idx = t;
        break;
      }
    }
    float mem511 = fmaf(Am, 511.0f * SPIKE_DT, -Bm);
    float spike  = (mem511 > 0.0f) ? ((float)idx * SPIKE_DT - 1.0f) : 1.0f;
    hbuf[v + 8 * half][oBase + r] = spike;
  }

  __syncthreads();

  // ------------------------- Layer 2 (wave 0) -------------------------
  // Uniform (scalar) predicate so inactive waves take a scalar branch and
  // never issue WMMA with EXEC != all-ones.
  const bool isWave0 = (__builtin_amdgcn_readfirstlane(wave) == 0);
  if (isWave0) {
    v8f accC = {};  // mask(h)   @ W2^T
    v8f accD = {};  // mask(h)*h @ W2^T
    for (int k = 0; k < 128; k += 4) {
      const int kk = k + 2 * half;
      float h0 = hbuf[r][kk];
      float h1 = hbuf[r][kk + 1];
      v2f am;  am.x = (h0 == 1.0f) ? 0.0f : 1.0f;
               am.y = (h1 == 1.0f) ? 0.0f : 1.0f;
      v2f ah;  ah.x = (h0 == 1.0f) ? 0.0f : h0;
               ah.y = (h1 == 1.0f) ? 0.0f : h1;
      v2f bf;  // N padded 10 -> 16 with zero rows of W2
      bf.x = (r < 10) ? W2[r * 128 + kk]     : 0.0f;
      bf.y = (r < 10) ? W2[r * 128 + kk + 1] : 0.0f;
      accC = __builtin_amdgcn_wmma_f32_16x16x4_f32(false, am, false, bf,
                                                   (short)0, accC, false, false);
      accD = __builtin_amdgcn_wmma_f32_16x16x4_f32(false, ah, false, bf,
                                                   (short)0, accD, false, false);
    }

#pragma unroll
    for (int v = 0; v < 8; ++v) {
      float Cv = accC[v];
      float Dv = accD[v];
      float res = __builtin_inff();
#pragma unroll 1
      for (int ks = 0; ks < 1024; ++ks) {
        float tau = 1.0f + (float)ks * SPIKE_DT;
        if (fmaf(Cv, tau, -Dv) > 20.0f) {
          res = tau;
          break;
        }
      }
      if (r < 10) {
        out[(bBase + v + 8 * half) * 10 + r] = res;
      }
    }
  }
}

extern "C" void kernel_launch(void* const* d_in, const int* in_sizes, int n_in,
                              void* d_out, int out_size, void* d_ws, size_t ws_size,
                              hipStream_t stream) {
  (void)in_sizes; (void)n_in; (void)d_ws; (void)ws_size; (void)out_size;
  const float* x  = (const float*)d_in[0];   // [128,800] f32
  const float* W1 = (const float*)d_in[1];   // [128,800] f32
  const float* W2 = (const float*)d_in[2];   // [10,128]  f32
  float* out = (float*)d_out;                // [128,10]  f32

  // 8 workgroups (one per 16-batch tile) x 256 threads (8 waves).
  spike_mlp_fused<<<8, 256, 0, stream>>>(x, W1, W2, out);
}